// L1_W_loss_16535624089853
// MI455X (gfx1250) — compile-verified
//
#include <hip/hip_runtime.h>
#include <math.h>
#include <stdint.h>

// ---------------------------------------------------------------------------
// Sinkhorn entropic OT on MI455X (gfx1250, wave32).
//   x,y: [16,3,512,512] f32.  48 independent (b,c) pairs, H=W=512.
//   Phase 1: C = x2 + y2 - 2*x.y^T via V_WMMA_F32_16X16X4_F32 (f32 matrix pipe
//            since C/eps multiplies any input error by 100 inside exp), with
//            double-buffered GLOBAL_LOAD_ASYNC_TO_LDS_B128 staging (ASYNCcnt)
//            overlapping DMA with the WMMA bursts.
//   Phase 2: 100 Sinkhorn iters; C (50MB) is L2-resident (192MB L2).
//   Phase 3: sum(exp(M)*C) with deterministic two-stage reduction.
// ---------------------------------------------------------------------------

typedef float v2f __attribute__((ext_vector_type(2)));
typedef float v8f __attribute__((ext_vector_type(8)));

#define NPAIRS   48
#define HDIM     512
#define NITER    100
#define INV_EPS  100.0f
#define EPS      0.01f
#define LOG_MU   (-6.23831946f)      /* log(1/512 + 1e-8) */
#define ERR_THR  1.6f                /* THRESH(0.1) * B(16): err summed over batch */
#define LDS_STR  68                  /* floats; 272B rows: 16B-aligned for async b128,
                                        4*lane mod 64 bank pattern -> conflict-free */

// ---- CDNA5 async global->LDS DMA (ASYNCcnt), GVS mode: mem = saddr+voff ----
__device__ __forceinline__ void async_load_b128(uint32_t lds_addr, uint32_t voff,
                                                const void* sbase) {
    asm volatile("global_load_async_to_lds_b128 %0, %1, %2"
                 :: "v"(lds_addr), "v"(voff), "s"(sbase) : "memory");
}
__device__ __forceinline__ void wait_asynccnt0() {
    asm volatile("s_wait_asynccnt 0x0" ::: "memory");
}
__device__ __forceinline__ uint32_t lds_u32(const void* p) {
    // generic LDS pointer: low 32 bits are the wave-relative LDS byte offset
    return (uint32_t)(uintptr_t)p;
}

// ---------------- row squared-norms: x2[p][i] = sum_w x[p][i][w]^2 ----------
__global__ void rownorms_kernel(const float* __restrict__ x,
                                const float* __restrict__ y,
                                float* __restrict__ x2, float* __restrict__ y2) {
    const int wave = threadIdx.x >> 5, lane = threadIdx.x & 31;
    const long R = (long)NPAIRS * HDIM;
    long id = (long)blockIdx.x * 8 + wave;
    const float* src = x; float* dst = x2; long row = id;
    if (id >= R) { src = y; dst = y2; row -= R; }
    const float* r = src + row * HDIM;
    float s = 0.0f;
    #pragma unroll
    for (int t = 0; t < 16; ++t) { float v = r[lane + t * 32]; s += v * v; }
    #pragma unroll
    for (int o = 16; o > 0; o >>= 1) s += __shfl_xor(s, o, 32);
    if (lane == 0) dst[row] = s;
}

// ---------------- C[p][i][j] = x2[i]+y2[j]-2*dot(x_i,y_j) via f32 WMMA ------
// Block: 256 thr = 8 waves (2x4), tile 32(M) x 64(N), K in 64-chunks,
// double-buffered LDS fed by async-to-LDS DMA.
__global__ void gemm_wmma_kernel(const float* __restrict__ x,
                                 const float* __restrict__ y,
                                 const float* __restrict__ x2,
                                 const float* __restrict__ y2,
                                 float* __restrict__ Cmat) {
    __shared__ float lds_buf[2][96 * LDS_STR];   // per buffer: A 32 rows + B 64 rows
    const int p       = blockIdx.z;
    const int rowBase = blockIdx.y * 32;
    const int colBase = blockIdx.x * 64;
    const int tid  = threadIdx.x;
    const int wave = tid >> 5, lane = tid & 31;
    const int waveM = wave >> 2, waveN = wave & 3;   // 2 x 4 wave grid
    const int half = lane >> 4, l15 = lane & 15;
    const float* xp = x + (size_t)p * HDIM * HDIM;
    const float* yp = y + (size_t)p * HDIM * HDIM;

    const uint32_t buf_bytes = 96 * LDS_STR * 4;
    const uint32_t xs0 = lds_u32(&lds_buf[0][0]);
    const uint32_t ys0 = xs0 + 32 * LDS_STR * 4;

    // per-thread staging slots (16B granules), global offsets in bytes
    const int srow = tid >> 4;             // 0..15
    const int sc4  = (tid & 15) * 16;      // byte column within 64-float row

    auto stage = [&](int kb, uint32_t bufsel) {   // kb in floats
        const uint32_t kbb = (uint32_t)kb * 4;
        const uint32_t bofs = bufsel * buf_bytes;
        #pragma unroll
        for (int pass = 0; pass < 2; ++pass) {    // A: 32 rows x 256B
            const int row = srow + pass * 16;
            async_load_b128(xs0 + bofs + (uint32_t)row * (LDS_STR * 4) + sc4,
                            (uint32_t)(rowBase + row) * (HDIM * 4) + kbb + sc4, xp);
        }
        #pragma unroll
        for (int pass = 0; pass < 4; ++pass) {    // B: 64 rows x 256B
            const int row = srow + pass * 16;
            async_load_b128(ys0 + bofs + (uint32_t)row * (LDS_STR * 4) + sc4,
                            (uint32_t)(colBase + row) * (HDIM * 4) + kbb + sc4, yp);
        }
    };

    v8f acc = {};
    stage(0, 0);
    wait_asynccnt0();
    __syncthreads();

    for (int ck = 0; ck < 8; ++ck) {
        if (ck < 7) stage((ck + 1) * 64, (uint32_t)((ck + 1) & 1));  // overlap DMA

        const float* xsb = &lds_buf[ck & 1][0];
        const float* ysb = &lds_buf[ck & 1][32 * LDS_STR];
        // 16 WMMAs of K=4. A 16x4 layout: lanes0-15 hold K0,K1; lanes16-31 K2,K3.
        #pragma unroll
        for (int kk = 0; kk < 64; kk += 4) {
            v2f a = *(const v2f*)&xsb[(waveM * 16 + l15) * LDS_STR + kk + half * 2];
            v2f b = *(const v2f*)&ysb[(waveN * 16 + l15) * LDS_STR + kk + half * 2];
            acc = __builtin_amdgcn_wmma_f32_16x16x4_f32(
                false, a, false, b, (short)0, acc, false, false);
        }
        wait_asynccnt0();   // next buffer's DMA complete (wave-local)
        __syncthreads();    // all waves done reading current buffer + DMA visible
    }

    // epilogue: D layout per ISA (VGPR g: lanes0-15 -> M=g, lanes16-31 -> M=g+8)
    float* Cp = Cmat + (size_t)p * HDIM * HDIM;
    const float* x2p = x2 + p * HDIM;
    const float* y2p = y2 + p * HDIM;
    const int col = colBase + waveN * 16 + l15;
    const float y2v = y2p[col];
    #pragma unroll
    for (int g = 0; g < 8; ++g) {
        const int row = rowBase + waveM * 16 + half * 8 + g;
        Cp[(size_t)row * HDIM + col] = x2p[row] + y2v - 2.0f * acc[g];
    }
}

// done flag for channel c at iteration `iter`: any prior err below threshold.
// Frozen iterations leave err_hist==0 (<thr) so done stays latched.
__device__ __forceinline__ bool channel_done(const float* eh, int c, int iter) {
    for (int t = 0; t < iter; ++t)
        if (eh[c * NITER + t] < ERR_THR) return true;
    return false;
}

// ------------- u update: u += eps*(log_mu - logsumexp_j M(u,v)) -------------
__global__ void sinkhorn_row_kernel(const float* __restrict__ Cmat,
                                    const float* __restrict__ u,
                                    const float* __restrict__ v,
                                    float* __restrict__ u_new,
                                    float* __restrict__ err_hist, int iter) {
    const int p  = blockIdx.x >> 6;
    const int rg = blockIdx.x & 63;
    const int c  = p % 3;
    if (channel_done(err_hist, c, iter)) return;

    __shared__ float vs[HDIM];
    __shared__ float perr[8];
    const float* vp = v + p * HDIM;
    for (int t = threadIdx.x; t < HDIM; t += 256) vs[t] = vp[t];
    __syncthreads();

    const int wave = threadIdx.x >> 5, lane = threadIdx.x & 31;
    const int i = rg * 8 + wave;
    const float* Crow = Cmat + (size_t)p * HDIM * HDIM + (size_t)i * HDIM;
    const float ui = u[p * HDIM + i];

    float m[16];
    float mx = -3.4e38f;
    #pragma unroll
    for (int t = 0; t < 16; ++t) {
        const int j = t * 32 + lane;
        m[t] = (ui + vs[j] - Crow[j]) * INV_EPS;
        mx = fmaxf(mx, m[t]);
    }
    #pragma unroll
    for (int o = 16; o > 0; o >>= 1) mx = fmaxf(mx, __shfl_xor(mx, o, 32));
    float s = 0.0f;
    #pragma unroll
    for (int t = 0; t < 16; ++t) s += __expf(m[t] - mx);
    #pragma unroll
    for (int o = 16; o > 0; o >>= 1) s += __shfl_xor(s, o, 32);
    const float lse = mx + __logf(s);
    const float un  = ui + EPS * (LOG_MU - lse);
    if (lane == 0) { u_new[p * HDIM + i] = un; perr[wave] = fabsf(un - ui); }
    __syncthreads();
    if (threadIdx.x == 0) {
        float e = 0.0f;
        #pragma unroll
        for (int w = 0; w < 8; ++w) e += perr[w];
        atomicAdd(&err_hist[c * NITER + iter], e);
    }
}

// ------------- v update (uses fresh u_new) + commit u_new -> u --------------
__global__ void sinkhorn_col_kernel(const float* __restrict__ Cmat,
                                    const float* __restrict__ u_new,
                                    float* __restrict__ u,
                                    float* __restrict__ v,
                                    const float* __restrict__ err_hist, int iter) {
    const int p  = blockIdx.x >> 4;
    const int cg = blockIdx.x & 15;
    const int c  = p % 3;
    if (channel_done(err_hist, c, iter)) return;

    __shared__ float us[HDIM];
    __shared__ float pmax[8][32];
    __shared__ float psum[8][32];
    const float* unp = u_new + p * HDIM;
    for (int t = threadIdx.x; t < HDIM; t += 256) us[t] = unp[t];
    __syncthreads();

    const int wave = threadIdx.x >> 5, lane = threadIdx.x & 31;
    const int j = cg * 32 + lane;
    const float* Cp = Cmat + (size_t)p * HDIM * HDIM;
    const float vj = v[p * HDIM + j];

    float mx = -3.4e38f;
    for (int r = 0; r < 64; ++r) {
        const int i = wave * 64 + r;
        const float val = (us[i] + vj - Cp[(size_t)i * HDIM + j]) * INV_EPS;
        mx = fmaxf(mx, val);
    }
    pmax[wave][lane] = mx;
    __syncthreads();
    float cmx = pmax[0][lane];
    #pragma unroll
    for (int w = 1; w < 8; ++w) cmx = fmaxf(cmx, pmax[w][lane]);

    float s = 0.0f;
    for (int r = 0; r < 64; ++r) {
        const int i = wave * 64 + r;
        const float val = (us[i] + vj - Cp[(size_t)i * HDIM + j]) * INV_EPS;
        s += __expf(val - cmx);
    }
    psum[wave][lane] = s;
    __syncthreads();
    if (wave == 0) {
        float tot = 0.0f;
        #pragma unroll
        for (int w = 0; w < 8; ++w) tot += psum[w][lane];
        const float lse = cmx + __logf(tot);
        v[p * HDIM + j] = vj + EPS * (LOG_MU - lse);
    }
    if (cg == 0) {  // commit u (u and u_new are distinct buffers)
        for (int t = threadIdx.x; t < HDIM; t += 256) u[p * HDIM + t] = us[t];
    }
}

// ------------- cost = sum exp(M(u,v)) * C, deterministic two-stage ----------
__global__ void cost_partial_kernel(const float* __restrict__ Cmat,
                                    const float* __restrict__ u,
                                    const float* __restrict__ v,
                                    float* __restrict__ partials) {
    const int p  = blockIdx.x >> 6;
    const int rg = blockIdx.x & 63;
    __shared__ float vs[HDIM];
    __shared__ float ps[8];
    const float* vp = v + p * HDIM;
    for (int t = threadIdx.x; t < HDIM; t += 256) vs[t] = vp[t];
    __syncthreads();
    const int wave = threadIdx.x >> 5, lane = threadIdx.x & 31;
    const int i = rg * 8 + wave;
    const float* Crow = Cmat + (size_t)p * HDIM * HDIM + (size_t)i * HDIM;
    const float ui = u[p * HDIM + i];
    float s = 0.0f;
    #pragma unroll
    for (int t = 0; t < 16; ++t) {
        const int j = t * 32 + lane;
        const float Cv = Crow[j];
        s += __expf((ui + vs[j] - Cv) * INV_EPS) * Cv;
    }
    #pragma unroll
    for (int o = 16; o > 0; o >>= 1) s += __shfl_xor(s, o, 32);
    if (lane == 0) ps[wave] = s;
    __syncthreads();
    if (threadIdx.x == 0) {
        float e = 0.0f;
        #pragma unroll
        for (int w = 0; w < 8; ++w) e += ps[w];
        partials[blockIdx.x] = e;
    }
}

__global__ void final_reduce_kernel(const float* __restrict__ partials,
                                    float* __restrict__ out, int n) {
    __shared__ float red[256];
    float s = 0.0f;
    for (int t = threadIdx.x; t < n; t += 256) s += partials[t];
    red[threadIdx.x] = s;
    __syncthreads();
    for (int o = 128; o > 0; o >>= 1) {
        if (threadIdx.x < o) red[threadIdx.x] += red[threadIdx.x + o];
        __syncthreads();
    }
    if (threadIdx.x == 0) out[0] = red[0];
}

// ---------------------------------------------------------------------------
extern "C" void kernel_launch(void* const* d_in, const int* in_sizes, int n_in,
                              void* d_out, int out_size, void* d_ws, size_t ws_size,
                              hipStream_t stream) {
    const float* x = (const float*)d_in[0];
    const float* y = (const float*)d_in[1];
    float* out = (float*)d_out;

    char* ws = (char*)d_ws;
    size_t off = 0;
    float* Cmat     = (float*)(ws + off); off += (size_t)NPAIRS * HDIM * HDIM * 4;
    float* x2       = (float*)(ws + off); off += (size_t)NPAIRS * HDIM * 4;
    float* y2       = (float*)(ws + off); off += (size_t)NPAIRS * HDIM * 4;
    float* u        = (float*)(ws + off); off += (size_t)NPAIRS * HDIM * 4;
    float* u_new    = (float*)(ws + off); off += (size_t)NPAIRS * HDIM * 4;
    float* v        = (float*)(ws + off); off += (size_t)NPAIRS * HDIM * 4;
    float* err_hist = (float*)(ws + off); off += 4096;
    float* partials = (float*)(ws + off); off += (size_t)NPAIRS * 64 * 4;

    hipMemsetAsync(u,        0, (size_t)NPAIRS * HDIM * 4, stream);
    hipMemsetAsync(v,        0, (size_t)NPAIRS * HDIM * 4, stream);
    hipMemsetAsync(err_hist, 0, 4096, stream);

    rownorms_kernel<<<(2 * NPAIRS * HDIM) / 8, 256, 0, stream>>>(x, y, x2, y2);
    gemm_wmma_kernel<<<dim3(HDIM / 64, HDIM / 32, NPAIRS), 256, 0, stream>>>(
        x, y, x2, y2, Cmat);

    for (int it = 0; it < NITER; ++it) {
        sinkhorn_row_kernel<<<NPAIRS * 64, 256, 0, stream>>>(Cmat, u, v, u_new,
                                                             err_hist, it);
        sinkhorn_col_kernel<<<NPAIRS * 16, 256, 0, stream>>>(Cmat, u_new, u, v,
                                                             err_hist, it);
    }

    cost_partial_kernel<<<NPAIRS * 64, 256, 0, stream>>>(Cmat, u, v, partials);
    final_reduce_kernel<<<1, 256, 0, stream>>>(partials, out, NPAIRS * 64);
}